// GenuineEmbedding_137438953634
// MI455X (gfx1250) — compile-verified
//
#include <hip/hip_runtime.h>
#include <math.h>

#define T_TOKENS 32768
#define D_DIM    1024
#define WAVES_PER_BLOCK 8
#define BLOCK    256
#define NBLOCKS  1024
#define ROW_F4   (D_DIM / 4)   // 256 float4 per token row

// Issue one token row (4KB) as 8 async b128 copies: each lane moves 16B per issue.
// Per CDNA5 ISA (GLOBAL_LOAD_ASYNC_TO_LDS, GVS mode):
//   the instruction offset is added to both the per-lane LDS address and the global address
// so the same lds-addr/global-offset VGPRs work for all 8 issues with increasing offset:.
__device__ __forceinline__ void async_copy_row(uint32_t lds_addr, uint32_t goff, uint64_t base)
{
    asm volatile("global_load_async_to_lds_b128 %0, %1, %2 offset:0"    :: "v"(lds_addr), "v"(goff), "s"(base) : "memory");
    asm volatile("global_load_async_to_lds_b128 %0, %1, %2 offset:512"  :: "v"(lds_addr), "v"(goff), "s"(base) : "memory");
    asm volatile("global_load_async_to_lds_b128 %0, %1, %2 offset:1024" :: "v"(lds_addr), "v"(goff), "s"(base) : "memory");
    asm volatile("global_load_async_to_lds_b128 %0, %1, %2 offset:1536" :: "v"(lds_addr), "v"(goff), "s"(base) : "memory");
    asm volatile("global_load_async_to_lds_b128 %0, %1, %2 offset:2048" :: "v"(lds_addr), "v"(goff), "s"(base) : "memory");
    asm volatile("global_load_async_to_lds_b128 %0, %1, %2 offset:2560" :: "v"(lds_addr), "v"(goff), "s"(base) : "memory");
    asm volatile("global_load_async_to_lds_b128 %0, %1, %2 offset:3072" :: "v"(lds_addr), "v"(goff), "s"(base) : "memory");
    asm volatile("global_load_async_to_lds_b128 %0, %1, %2 offset:3584" :: "v"(lds_addr), "v"(goff), "s"(base) : "memory");
}

__device__ __forceinline__ float comp_elem(float r, float tp, float sc, float sh, float& acc)
{
    float a = r * tp;
    float c = __cosf(a);           // v_cos_f32 path (trans unit)
    float s = __sinf(a);           // v_sin_f32 path
    float v = c + c * s + s * s;   // c*(1+s) + s*s
    v = v * sc + sh;
    acc += v * v;
    return v;
}

__global__ __launch_bounds__(BLOCK) void resonance_encode_kernel(
    const float* __restrict__ token_values,
    const float* __restrict__ positions,
    const float* __restrict__ resonances,
    const float* __restrict__ scales,
    const float* __restrict__ shifts,
    const float* __restrict__ energy_norm,
    float* __restrict__ out)
{
    __shared__ float  s_scales[D_DIM];                       // 4 KB
    __shared__ float  s_shifts[D_DIM];                       // 4 KB
    __shared__ float4 s_buf[2][WAVES_PER_BLOCK * ROW_F4];    // 64 KB double buffer

    const int tid = threadIdx.x;

    // Stage scales/shifts once per block (256 threads x 1 float4 each covers 1024 floats).
    ((float4*)s_scales)[tid] = ((const float4*)scales)[tid];
    ((float4*)s_shifts)[tid] = ((const float4*)shifts)[tid];
    __syncthreads();

    const int lane  = tid & 31;
    const int wave  = tid >> 5;
    const int gwave = blockIdx.x * WAVES_PER_BLOCK + wave;
    const int nwaves = NBLOCKS * WAVES_PER_BLOCK;

    const uint64_t res_base = (uint64_t)resonances;
    // Flat->LDS: low 32 bits of the flat shared address are the LDS byte offset.
    const uint32_t lds0 = (uint32_t)(uint64_t)(&s_buf[0][wave * ROW_F4]) + (uint32_t)(lane * 16);
    const uint32_t lds1 = (uint32_t)(uint64_t)(&s_buf[1][wave * ROW_F4]) + (uint32_t)(lane * 16);

    const float en = energy_norm[0];

    int t = gwave;
    if (t < T_TOKENS)
        async_copy_row(lds0, (uint32_t)(t * (D_DIM * 4) + lane * 16), res_base);

    int buf = 0;
    for (; t < T_TOKENS; t += nwaves) {
        const int tn = t + nwaves;
        if (tn < T_TOKENS) {
            // Prefetch next row into the other buffer, then wait until only those
            // 8 new async ops remain outstanding (in-order completion => tile t done).
            async_copy_row(buf ? lds0 : lds1,
                           (uint32_t)(tn * (D_DIM * 4) + lane * 16), res_base);
            asm volatile("s_wait_asynccnt 8" ::: "memory");
        } else {
            asm volatile("s_wait_asynccnt 0" ::: "memory");
        }

        const float tp = token_values[t] + positions[t];   // wave-uniform scalar
        const float4* rowv = &s_buf[buf][wave * ROW_F4];

        float  sumsq = 0.0f;
        float4 comp[8];
        #pragma unroll
        for (int j = 0; j < 8; ++j) {
            const int idx = j * 32 + lane;                 // float4 index within row
            float4 r  = rowv[idx];                         // ds_load_b128
            float4 sc = ((const float4*)s_scales)[idx];
            float4 sh = ((const float4*)s_shifts)[idx];
            float4 v;
            v.x = comp_elem(r.x, tp, sc.x, sh.x, sumsq);
            v.y = comp_elem(r.y, tp, sc.y, sh.y, sumsq);
            v.z = comp_elem(r.z, tp, sc.z, sh.z, sumsq);
            v.w = comp_elem(r.w, tp, sc.w, sh.w, sumsq);
            comp[j] = v;
        }

        // wave32 butterfly reduction for the row's sum of squares
        #pragma unroll
        for (int off = 16; off > 0; off >>= 1)
            sumsq += __shfl_xor(sumsq, off, 32);

        const float nrm = sqrtf(sumsq);
        const float f   = (nrm > 0.0f) ? (en / nrm) : 1.0f;

        float4* orow = (float4*)(out + (size_t)t * D_DIM);
        #pragma unroll
        for (int j = 0; j < 8; ++j) {
            float4 v = comp[j];
            v.x *= f; v.y *= f; v.z *= f; v.w *= f;
            orow[j * 32 + lane] = v;                       // global_store_b128
        }
        buf ^= 1;
    }
}

extern "C" void kernel_launch(void* const* d_in, const int* in_sizes, int n_in,
                              void* d_out, int out_size, void* d_ws, size_t ws_size,
                              hipStream_t stream)
{
    (void)in_sizes; (void)n_in; (void)out_size; (void)d_ws; (void)ws_size;
    const float* token_values = (const float*)d_in[0];
    const float* positions    = (const float*)d_in[1];
    const float* resonances   = (const float*)d_in[2];
    const float* scales       = (const float*)d_in[3];
    const float* shifts       = (const float*)d_in[4];
    const float* energy_norm  = (const float*)d_in[5];
    float* out = (float*)d_out;

    resonance_encode_kernel<<<NBLOCKS, BLOCK, 0, stream>>>(
        token_values, positions, resonances, scales, shifts, energy_norm, out);
}